// ReverseDiffusionModel_1108101562859
// MI455X (gfx1250) — compile-verified
//
#include <hip/hip_runtime.h>
#include <hip/hip_bf16.h>
#include <math.h>

// Problem constants (from reference)
#define Bz 16
#define Lz 128
#define Nz 50000
#define Ez 128
#define Hz 64
#define HEADSz 2
#define Dz 64
#define QKVz 384      // 3 * HEADS * DIM_HEAD
#define C1z 320       // E + 3*H
#define SCALEf 0.125f // 64^-0.5

typedef float v2f __attribute__((ext_vector_type(2)));
typedef float v8f __attribute__((ext_vector_type(8)));

// ---------------------------------------------------------------------------
// K1: fused gather + mask + QKV projection:  qkv[b][o][l] = sum_e Wqkv[o][e]*xm
// ---------------------------------------------------------------------------
__global__ __launch_bounds__(256) void k_qkv(const int* __restrict__ path,
                                             const int* __restrict__ masks,
                                             const float* __restrict__ phi,
                                             const float* __restrict__ Wqkv,
                                             float* __restrict__ qkv) {
  int idx = blockIdx.x * blockDim.x + threadIdx.x;
  if (idx >= Bz * QKVz * Lz) return;
  int l = idx % Lz;
  int o = (idx / Lz) % QKVz;
  int b = idx / (Lz * QKVz);
  float acc = 0.f;
  if (!masks[b * Lz + l]) {
    int node = path[b * Lz + l];
    const float* x = phi + (size_t)node * Ez;
    const float* w = Wqkv + o * Ez;
#pragma unroll 4
    for (int e = 0; e < Ez; ++e) acc += w[e] * x[e];
  }
  qkv[idx] = acc;
}

// ---------------------------------------------------------------------------
// K2: softmax over l for k (with -1e15 mask fill). One block per (b, hd) row.
// ---------------------------------------------------------------------------
__global__ __launch_bounds__(128) void k_softmax(const int* __restrict__ masks,
                                                 const float* __restrict__ qkv,
                                                 float* __restrict__ ksoft) {
  __shared__ float red[Lz];
  int row = blockIdx.x;     // b*128 + hd
  int b = row >> 7;
  int hd = row & 127;
  int l = threadIdx.x;
  float kv = masks[b * Lz + l] ? -1e15f
                               : qkv[((b * QKVz) + Ez + hd) * Lz + l];
  red[l] = kv;
  __syncthreads();
  for (int s = 64; s > 0; s >>= 1) {
    if (l < s) red[l] = fmaxf(red[l], red[l + s]);
    __syncthreads();
  }
  float mx = red[0];
  __syncthreads();
  float ev = expf(kv - mx);
  red[l] = ev;
  __syncthreads();
  for (int s = 64; s > 0; s >>= 1) {
    if (l < s) red[l] += red[l + s];
    __syncthreads();
  }
  ksoft[row * Lz + l] = ev / red[0];
}

// ---------------------------------------------------------------------------
// K3: context[b][h][d][e] = sum_l ksoft[b][h,d][l] * v[b][h,e][l]
// ---------------------------------------------------------------------------
__global__ __launch_bounds__(256) void k_context(const float* __restrict__ ksoft,
                                                 const float* __restrict__ qkv,
                                                 float* __restrict__ ctx) {
  int idx = blockIdx.x * blockDim.x + threadIdx.x;
  if (idx >= Bz * HEADSz * Dz * Dz) return;
  int e = idx & 63;
  int d = (idx >> 6) & 63;
  int h = (idx >> 12) & 1;
  int b = idx >> 13;
  const float* kr = ksoft + ((b * Ez) + h * Dz + d) * Lz;
  const float* vr = qkv + ((b * QKVz) + 2 * Ez + h * Dz + e) * Lz;
  float acc = 0.f;
#pragma unroll 4
  for (int l = 0; l < Lz; ++l) acc += kr[l] * vr[l];
  ctx[idx] = acc;
}

// ---------------------------------------------------------------------------
// K4: pre[b][o][l] = SCALE * sum_e ctx[b][h][d][e] * q[b][h,e][l]   (o=h*64+d)
// ---------------------------------------------------------------------------
__global__ __launch_bounds__(256) void k_attnpre(const float* __restrict__ ctx,
                                                 const float* __restrict__ qkv,
                                                 float* __restrict__ pre) {
  int idx = blockIdx.x * blockDim.x + threadIdx.x;
  if (idx >= Bz * Ez * Lz) return;
  int l = idx & 127;
  int o = (idx >> 7) & 127;
  int b = idx >> 14;
  int h = o >> 6, d = o & 63;
  const float* cr = ctx + (((b * HEADSz) + h) * Dz + d) * Dz;
  const float* qb = qkv + (b * QKVz + h * Dz) * Lz + l;
  float acc = 0.f;
#pragma unroll 4
  for (int e = 0; e < Dz; ++e) acc += cr[e] * qb[e * Lz];
  pre[idx] = acc * SCALEf;
}

// ---------------------------------------------------------------------------
// K5: xfeat[b][l][0:128] = Wout @ pre + bout
// ---------------------------------------------------------------------------
__global__ __launch_bounds__(256) void k_wout(const float* __restrict__ pre,
                                              const float* __restrict__ Wout,
                                              const float* __restrict__ bout,
                                              float* __restrict__ xfeat) {
  int idx = blockIdx.x * blockDim.x + threadIdx.x;
  if (idx >= Bz * Lz * Ez) return;
  int e = idx & 127;
  int l = (idx >> 7) & 127;
  int b = idx >> 14;
  const float* w = Wout + e * Ez;
  const float* p = pre + (b * Ez) * Lz + l;
  float acc = bout[e];
#pragma unroll 4
  for (int o = 0; o < Ez; ++o) acc += w[o] * p[o * Lz];
  xfeat[(b * Lz + l) * C1z + e] = acc;
}

// ---------------------------------------------------------------------------
// K6: xfeat[b][l][128:320] = [t_emb(64) | od_emb(64) | spatial(64)]
// ---------------------------------------------------------------------------
__global__ __launch_bounds__(256) void k_misc(const int* __restrict__ path,
                                              const int* __restrict__ masks,
                                              const float* __restrict__ t,
                                              const int* __restrict__ ori,
                                              const int* __restrict__ dstn,
                                              const float* __restrict__ pos,
                                              const float* __restrict__ dst_pos,
                                              const float* __restrict__ phi,
                                              const float* __restrict__ Wt,
                                              const float* __restrict__ bt,
                                              const float* __restrict__ Wod,
                                              const float* __restrict__ bod,
                                              const float* __restrict__ Wsp,
                                              const float* __restrict__ bsp,
                                              float* __restrict__ xfeat) {
  int idx = blockIdx.x * blockDim.x + threadIdx.x;
  if (idx >= Bz * Lz * 192) return;
  int c = idx % 192;
  int l = (idx / 192) % Lz;
  int b = idx / (192 * Lz);
  float val;
  if (c < 64) {  // t_emb
    int h = c;
    val = t[b] * Wt[h] + bt[h];
  } else if (c < 128) {  // od_emb
    int h = c - 64;
    const float* po = phi + (size_t)ori[b] * Ez;
    const float* pd = phi + (size_t)dstn[b] * Ez;
    const float* w = Wod + h * (2 * Ez);
    float acc = bod[h];
#pragma unroll 4
    for (int j = 0; j < Ez; ++j) acc += po[j] * w[j];
#pragma unroll 4
    for (int j = 0; j < Ez; ++j) acc += pd[j] * w[Ez + j];
    val = acc;
  } else {  // spatial
    int h = c - 128;
    float sd = 0.f, sdir = 0.f;
    if (!masks[b * Lz + l]) {
      int node = path[b * Lz + l];
      float dx = dst_pos[b * 2 + 0] - pos[(size_t)node * 2 + 0];
      float dy = dst_pos[b * 2 + 1] - pos[(size_t)node * 2 + 1];
      float dist = sqrtf(dx * dx + dy * dy);
      sdir = (dist > 1e-6f) ? dx / fmaxf(dist, 1e-6f) : 0.f;
      sd = dist;
    }
    val = sd * Wsp[h * 2 + 0] + sdir * Wsp[h * 2 + 1] + bsp[h];
  }
  xfeat[(b * Lz + l) * C1z + 128 + c] = val;
}

// ---------------------------------------------------------------------------
// K7: conv1 (320 -> 64, k=3, SAME) + ReLU
// ---------------------------------------------------------------------------
__global__ __launch_bounds__(256) void k_conv1(const float* __restrict__ xfeat,
                                               const float* __restrict__ Wc1,
                                               const float* __restrict__ bc1,
                                               float* __restrict__ y1) {
  int idx = blockIdx.x * blockDim.x + threadIdx.x;
  if (idx >= Bz * Lz * Hz) return;
  int h = idx & 63;
  int l = (idx >> 6) & 127;
  int b = idx >> 13;
  float acc = bc1[h];
  for (int k = 0; k < 3; ++k) {
    int ll = l + k - 1;
    if ((unsigned)ll >= (unsigned)Lz) continue;
    const float* xr = xfeat + (b * Lz + ll) * C1z;
    const float* wr = Wc1 + h * C1z * 3 + k;
#pragma unroll 4
    for (int c = 0; c < C1z; ++c) acc += wr[c * 3] * xr[c];
  }
  y1[idx] = fmaxf(acc, 0.f);
}

// ---------------------------------------------------------------------------
// K8: conv2 (64 -> 64, k=3, SAME) + ReLU
// ---------------------------------------------------------------------------
__global__ __launch_bounds__(256) void k_conv2(const float* __restrict__ y1,
                                               const float* __restrict__ Wc2,
                                               const float* __restrict__ bc2,
                                               float* __restrict__ xfin) {
  int idx = blockIdx.x * blockDim.x + threadIdx.x;
  if (idx >= Bz * Lz * Hz) return;
  int h = idx & 63;
  int l = (idx >> 6) & 127;
  int b = idx >> 13;
  float acc = bc2[h];
  for (int k = 0; k < 3; ++k) {
    int ll = l + k - 1;
    if ((unsigned)ll >= (unsigned)Lz) continue;
    const float* xr = y1 + (b * Lz + ll) * Hz;
    const float* wr = Wc2 + h * Hz * 3 + k;
#pragma unroll 4
    for (int c = 0; c < Hz; ++c) acc += wr[c * 3] * xr[c];
  }
  xfin[idx] = fmaxf(acc, 0.f);
}

// ---------------------------------------------------------------------------
// K9: logits = xfin(2048x64) @ Wfc^T(64x50000) + bfc using V_WMMA_F32_16X16X4
// Each wave: 16x64 output tile (4 accumulators). Block of 8 waves: 32x256.
// ---------------------------------------------------------------------------
__global__ __launch_bounds__(256) void k_logits(const float* __restrict__ X,
                                                const float* __restrict__ W,
                                                const float* __restrict__ bias,
                                                float* __restrict__ out) {
  int wave = threadIdx.x >> 5;  // 0..7
  int lane = threadIdx.x & 31;
  int waveM = wave >> 2;        // 0..1
  int waveN = wave & 3;         // 0..3
  int m0 = blockIdx.y * 32 + waveM * 16;
  int n0 = blockIdx.x * 256 + waveN * 64;

  int lm = lane & 15;
  int hi = lane >> 4;  // lanes 16-31 hold K+2,K+3 halves of the fragments
  int row = m0 + lm;

  v8f acc[4] = {v8f{}, v8f{}, v8f{}, v8f{}};

  const float* Xrow = X + row * Hz + hi * 2;
  // Pre-clamp B columns so EXEC stays all-ones at every WMMA (garbage columns
  // are never stored).
  int colc[4];
#pragma unroll
  for (int g = 0; g < 4; ++g) {
    int col = n0 + g * 16 + lm;
    colc[g] = col < Nz ? col : (Nz - 1);
  }

  for (int kb = 0; kb < Hz; kb += 4) {
    v2f a;
    a.x = Xrow[kb];
    a.y = Xrow[kb + 1];
#pragma unroll
    for (int g = 0; g < 4; ++g) {
      const float* Wrow = W + (size_t)colc[g] * Hz + kb + hi * 2;
      v2f bf;
      bf.x = Wrow[0];
      bf.y = Wrow[1];
      acc[g] = __builtin_amdgcn_wmma_f32_16x16x4_f32(
          false, a, false, bf, (short)0, acc[g], false, false);
    }
  }

  int mBase = m0 + hi * 8;
#pragma unroll
  for (int g = 0; g < 4; ++g) {
    int col = n0 + g * 16 + lm;
    if (col < Nz) {
      float bv = bias[col];
#pragma unroll
      for (int r = 0; r < 8; ++r) {
        out[(size_t)(mBase + r) * Nz + col] = acc[g][r] + bv;
      }
    }
  }
}

// ---------------------------------------------------------------------------
extern "C" void kernel_launch(void* const* d_in, const int* in_sizes, int n_in,
                              void* d_out, int out_size, void* d_ws, size_t ws_size,
                              hipStream_t stream) {
  const int* path = (const int*)d_in[0];
  const float* t = (const float*)d_in[1];
  const int* ori = (const int*)d_in[2];
  const int* dstn = (const int*)d_in[3];
  const int* masks = (const int*)d_in[4];
  const float* pos = (const float*)d_in[5];
  const float* dst_pos = (const float*)d_in[6];
  const float* phi = (const float*)d_in[7];
  const float* Wt = (const float*)d_in[8];
  const float* bt = (const float*)d_in[9];
  const float* Wod = (const float*)d_in[10];
  const float* bod = (const float*)d_in[11];
  const float* Wsp = (const float*)d_in[12];
  const float* bsp = (const float*)d_in[13];
  const float* Wqkv = (const float*)d_in[14];
  const float* Wout = (const float*)d_in[15];
  const float* bout = (const float*)d_in[16];
  const float* Wc1 = (const float*)d_in[17];
  const float* bc1 = (const float*)d_in[18];
  const float* Wc2 = (const float*)d_in[19];
  const float* bc2 = (const float*)d_in[20];
  const float* Wfc = (const float*)d_in[21];
  const float* bfc = (const float*)d_in[22];
  float* out = (float*)d_out;
  float* ws = (float*)d_ws;

  // workspace layout (floats)
  float* qkv   = ws;                          // 16*384*128 = 786432
  float* ksoft = qkv + Bz * QKVz * Lz;        // 16*128*128 = 262144
  float* ctx   = ksoft + Bz * Ez * Lz;        // 16*2*64*64 = 131072
  float* pre   = ctx + Bz * HEADSz * Dz * Dz; // 262144
  float* xfeat = pre + Bz * Ez * Lz;          // 16*128*320 = 655360
  float* y1    = xfeat + Bz * Lz * C1z;       // 131072
  float* xfin  = y1 + Bz * Lz * Hz;           // 131072

  k_qkv<<<(Bz * QKVz * Lz + 255) / 256, 256, 0, stream>>>(path, masks, phi, Wqkv, qkv);
  k_softmax<<<Bz * Ez, Lz, 0, stream>>>(masks, qkv, ksoft);
  k_context<<<(Bz * HEADSz * Dz * Dz + 255) / 256, 256, 0, stream>>>(ksoft, qkv, ctx);
  k_attnpre<<<(Bz * Ez * Lz + 255) / 256, 256, 0, stream>>>(ctx, qkv, pre);
  k_wout<<<(Bz * Lz * Ez + 255) / 256, 256, 0, stream>>>(pre, Wout, bout, xfeat);
  k_misc<<<(Bz * Lz * 192 + 255) / 256, 256, 0, stream>>>(
      path, masks, t, ori, dstn, pos, dst_pos, phi, Wt, bt, Wod, bod, Wsp, bsp, xfeat);
  k_conv1<<<(Bz * Lz * Hz + 255) / 256, 256, 0, stream>>>(xfeat, Wc1, bc1, y1);
  k_conv2<<<(Bz * Lz * Hz + 255) / 256, 256, 0, stream>>>(y1, Wc2, bc2, xfin);

  dim3 g((Nz + 255) / 256, (Bz * Lz) / 32);
  k_logits<<<g, 256, 0, stream>>>(xfin, Wfc, bfc, out);
}